// mvPAM_Module_unfold_6871947674110
// MI455X (gfx1250) — compile-verified
//
#include <hip/hip_runtime.h>
#include <hip/hip_bf16.h>

typedef __attribute__((ext_vector_type(2))) float v2f;
typedef __attribute__((ext_vector_type(8))) float v8f;

namespace cfg {
constexpr int B   = 2;
constexpr int C   = 128;
constexpr int CQ  = 16;
constexpr int H   = 96;
constexpr int W   = 96;
constexpr int P   = H * W;        // 9216
constexpr int PAD = 3;
constexpr int PW  = W + 2 * PAD;  // 102 (padded width/height)
constexpr int PP  = PW * PW;      // 10404
constexpr int VIEW = 7;
constexpr int V2  = VIEW * VIEW;  // 49
constexpr int CO  = CQ + CQ + C;  // 160 fused output channels
constexpr int MT  = CO / 16;      // 10 row tiles (0:q, 1:k, 2..9:v)
constexpr int NT  = P / 16;       // 576 col tiles
}

// ---------------------------------------------------------------------------
// Kernel 1: fused QKV 1x1-conv GEMM, register-blocked over M.
// One wave32 per (batch, 16-pixel column tile); the wave computes all 10
// 16x16 row tiles (160 output channels), so the B operand (x) is loaded once
// per k-step and reused by 10 back-to-back V_WMMA_F32_16X16X4_F32 ops
// (fp32 precision preserved vs the fp32 reference; 10x less x traffic than
// tile-per-wave). Epilogue adds bias and routes q -> packed (B,16,P);
// k,v -> zero-padded (102x102) planes so attention needs no boundary checks.
// ---------------------------------------------------------------------------
__global__ __launch_bounds__(256) void qkv_gemm_wmma(
    const float* __restrict__ x,
    const float* __restrict__ Wq, const float* __restrict__ bq,
    const float* __restrict__ Wk, const float* __restrict__ bk,
    const float* __restrict__ Wv, const float* __restrict__ bv,
    float* __restrict__ qws, float* __restrict__ kpad, float* __restrict__ vpad)
{
  using namespace cfg;
  const int lane = threadIdx.x & 31;
  const int wid  = (blockIdx.x * blockDim.x + threadIdx.x) >> 5; // 0..B*NT-1
  const int b    = wid / NT;
  const int tn   = wid % NT;

  const float* xb = x + (size_t)b * C * P;
  const int hi   = lane >> 4;          // 0: K pair {0,1}   1: K pair {2,3}
  const int kofs = hi * 2;
  const int row  = lane & 15;
  const int n    = tn * 16 + row;      // this lane's output column (pixel)

  // Per-lane A-matrix row pointers for the 10 M tiles (wave-uniform routing:
  // 16-row tiles align exactly with the q/k/v channel split).
  const float* wrow[MT];
#pragma unroll
  for (int t = 0; t < MT; ++t) {
    const int m = t * 16 + row;
    wrow[t] = (m < CQ)      ? Wq + (size_t)m * C
            : (m < 2 * CQ)  ? Wk + (size_t)(m - CQ) * C
                            : Wv + (size_t)(m - 2 * CQ) * C;
  }

  v8f acc[MT];
#pragma unroll
  for (int t = 0; t < MT; ++t) acc[t] = (v8f){};

  for (int k0 = 0; k0 < C; k0 += 4) {
    // B 4x16 f32: VGPR0 = row (k0 | k0+2), VGPR1 = row (k0+1 | k0+3); loaded once.
    v2f bm;
    bm.x = xb[(size_t)(k0 + kofs) * P + n];
    bm.y = xb[(size_t)(k0 + kofs + 1) * P + n];
    if (k0 + 8 < C)  // prefetch next-next k-step rows (global_prefetch_b8)
      __builtin_prefetch(xb + (size_t)(k0 + 8 + kofs) * P + n, 0, 1);
#pragma unroll
    for (int t = 0; t < MT; ++t) {
      // A 16x4 f32: lane<16 holds K=k0,k0+1 of its row; lane>=16 K=k0+2,k0+3.
      v2f a = *(const v2f*)(wrow[t] + k0 + kofs);
      acc[t] = __builtin_amdgcn_wmma_f32_16x16x4_f32(
          /*neg_a=*/false, a, /*neg_b=*/false, bm,
          /*c_mod=*/(short)0, acc[t], /*reuse_a=*/false, /*reuse_b=*/false);
    }
  }

  const int py = n / W, px = n % W;
  const size_t ppix = (size_t)(py + PAD) * PW + (px + PAD);
#pragma unroll
  for (int t = 0; t < MT; ++t) {
#pragma unroll
    for (int r = 0; r < 8; ++r) {
      const int m = t * 16 + r + hi * 8;  // C/D layout: VGPR r = M=r (lo) / M=r+8 (hi)
      float val = acc[t][r];
      if (m < CQ) {
        val += bq[m];
        qws[((size_t)b * CQ + m) * P + n] = val;
      } else if (m < 2 * CQ) {
        val += bk[m - CQ];
        kpad[((size_t)b * CQ + (m - CQ)) * PP + ppix] = val;
      } else {
        val += bv[m - 2 * CQ];
        vpad[((size_t)b * C + (m - 2 * CQ)) * PP + ppix] = val;
      }
    }
  }
}

// ---------------------------------------------------------------------------
// Kernel 2: per-pixel energy (49 taps x 16-ch dot) + softmax -> attn[b][49][p].
// c-outer loop: each channel's 7x7 window read is coalesced across lanes
// (consecutive px) and hits L2 (k planes are 1.3 MB total).
// ---------------------------------------------------------------------------
__global__ __launch_bounds__(256) void attn_softmax(
    const float* __restrict__ qws, const float* __restrict__ kpad,
    float* __restrict__ attn)
{
  using namespace cfg;
  const int tid = blockIdx.x * blockDim.x + threadIdx.x;  // grid sized exactly B*P
  const int b = tid / P, p = tid % P;
  const int py = p / W, px = p % W;

  float e[V2];
#pragma unroll
  for (int i = 0; i < V2; ++i) e[i] = 0.f;

  for (int c = 0; c < CQ; ++c) {
    const float qv = qws[((size_t)b * CQ + c) * P + p];
    const float* kc = kpad + ((size_t)b * CQ + c) * PP + (size_t)py * PW + px;
#pragma unroll
    for (int dy = 0; dy < VIEW; ++dy)
#pragma unroll
      for (int dx = 0; dx < VIEW; ++dx)
        e[dy * VIEW + dx] = fmaf(qv, kc[dy * PW + dx], e[dy * VIEW + dx]);
  }

  float emax = e[0];
#pragma unroll
  for (int i = 1; i < V2; ++i) emax = fmaxf(emax, e[i]);
  float s = 0.f;
#pragma unroll
  for (int i = 0; i < V2; ++i) { e[i] = __expf(e[i] - emax); s += e[i]; }
  const float inv = 1.f / s;
#pragma unroll
  for (int i = 0; i < V2; ++i)
    attn[((size_t)b * V2 + i) * P + p] = e[i] * inv;
}

// ---------------------------------------------------------------------------
// Kernel 3: out[b,c,p] = gamma * sum_k attn[b,k,p] * vpad[b,c,p+off(k)] + x.
// One thread per output element; attn reads and vpad window reads are
// coalesced across consecutive p; vpad (10.7 MB) is fully L2 resident.
// ---------------------------------------------------------------------------
__global__ __launch_bounds__(256) void attn_apply(
    const float* __restrict__ attn, const float* __restrict__ vpad,
    const float* __restrict__ x, const float* __restrict__ gamma,
    float* __restrict__ out)
{
  using namespace cfg;
  const size_t idx = (size_t)blockIdx.x * blockDim.x + threadIdx.x; // exact B*C*P
  const int p = (int)(idx % P);
  const int c = (int)((idx / P) % C);
  const int b = (int)(idx / ((size_t)C * P));
  const int py = p / W, px = p % W;

  const float* vc = vpad + ((size_t)b * C + c) * PP + (size_t)py * PW + px;
  const float* ap = attn + (size_t)b * V2 * P + p;

  float s = 0.f;
#pragma unroll
  for (int dy = 0; dy < VIEW; ++dy)
#pragma unroll
    for (int dx = 0; dx < VIEW; ++dx)
      s = fmaf(ap[(size_t)(dy * VIEW + dx) * P], vc[dy * PW + dx], s);

  out[idx] = gamma[0] * s + x[idx];
}

// ---------------------------------------------------------------------------
extern "C" void kernel_launch(void* const* d_in, const int* in_sizes, int n_in,
                              void* d_out, int out_size, void* d_ws, size_t ws_size,
                              hipStream_t stream)
{
  using namespace cfg;
  const float* x     = (const float*)d_in[0];
  const float* Wq    = (const float*)d_in[1];
  const float* bq    = (const float*)d_in[2];
  const float* Wk    = (const float*)d_in[3];
  const float* bk    = (const float*)d_in[4];
  const float* Wv    = (const float*)d_in[5];
  const float* bv    = (const float*)d_in[6];
  const float* gamma = (const float*)d_in[7];
  float* out = (float*)d_out;

  // Workspace layout (floats): q | kpad | vpad | attn  (~16.8 MB total)
  float* qws  = (float*)d_ws;
  float* kpad = qws  + (size_t)B * CQ * P;
  float* vpad = kpad + (size_t)B * CQ * PP;
  float* attn = vpad + (size_t)B * C  * PP;

  // Zero the padded k/v planes each call (graph-capture-safe stream op);
  // interiors are overwritten by the GEMM kernel.
  hipMemsetAsync(kpad, 0,
                 ((size_t)B * CQ * PP + (size_t)B * C * PP) * sizeof(float),
                 stream);

  const int totalWaves = B * NT;                 // 1152 waves -> 144 blocks
  qkv_gemm_wmma<<<totalWaves / 8, 256, 0, stream>>>(
      x, Wq, bq, Wk, bk, Wv, bv, qws, kpad, vpad);

  attn_softmax<<<(B * P) / 256, 256, 0, stream>>>(qws, kpad, attn);

  attn_apply<<<(int)(((size_t)B * C * P) / 256), 256, 0, stream>>>(
      attn, vpad, x, gamma, out);
}